// VmapConvAttention_14336600834787
// MI455X (gfx1250) — compile-verified
//
#include <hip/hip_runtime.h>

#define N_NODES  50000
#define N_PAIRS  800000
#define NHEADS   4
#define HEAD_DIM 64
#define NFEAT    256   // NHEADS * HEAD_DIM

typedef __attribute__((ext_vector_type(2))) float v2f;
typedef __attribute__((ext_vector_type(4))) float v4f;
typedef __attribute__((ext_vector_type(8))) float v8f;

// ---------------------------------------------------------------------------
// Kernel 1: per-head QKV projection with f32 WMMA (16x16x4).
// One wave computes a 16-node x 16-outdim tile for one head, for q,k,v at once
// (A fragment reused across 3 B fragments -> 48 v_wmma per wave).
// Waves: (50000/16) mtiles * 4 heads * 4 ntiles = 50000 -> 6250 blocks exact,
// so EXEC is all-1s everywhere (WMMA requirement), no guards needed.
// A[m][k] = x[node0+m][head*64 + k]
// B[k][n] = W[head][ntile*16 + n][k]      (W is (H, out, in); nn.Linear W.T)
// Fragment layout per CDNA5 ISA 7.12.2 (32-bit A 16x4, B mirrored with N):
//   lane<16:  row/col = lane,    v0 -> K=kb+0, v1 -> K=kb+1
//   lane>=16: row/col = lane-16, v0 -> K=kb+2, v1 -> K=kb+3
// ---------------------------------------------------------------------------
__global__ void __launch_bounds__(256)
qkv_proj_wmma(const float* __restrict__ x,
              const float* __restrict__ Wq,
              const float* __restrict__ Wk,
              const float* __restrict__ Wv,
              float* __restrict__ q,
              float* __restrict__ k,
              float* __restrict__ v)
{
    const int lane  = threadIdx.x & 31;
    const int wave  = blockIdx.x * 8 + (threadIdx.x >> 5);  // 0..49999
    const int ntile = wave & 3;
    const int head  = (wave >> 2) & 3;
    const int mtile = wave >> 4;                            // 0..3124

    const int r  = lane & 15;
    const int kh = lane >> 4;       // 0 or 1: which K-pair within a K=4 step

    const float* xrow = x + (size_t)(mtile * 16 + r) * NFEAT + head * HEAD_DIM;
    const size_t woff = (size_t)head * HEAD_DIM * HEAD_DIM
                      + (size_t)(ntile * 16 + r) * HEAD_DIM;
    const float* wq = Wq + woff;
    const float* wk = Wk + woff;
    const float* wv = Wv + woff;

    v8f cq = {};
    v8f ck = {};
    v8f cv = {};

#pragma unroll
    for (int kk = 0; kk < 16; ++kk) {              // K = 64 in steps of 4
        const int kb = kk * 4 + kh * 2;
        v2f a, bq, bk, bv;
        a.x  = xrow[kb]; a.y  = xrow[kb + 1];
        bq.x = wq[kb];   bq.y = wq[kb + 1];
        bk.x = wk[kb];   bk.y = wk[kb + 1];
        bv.x = wv[kb];   bv.y = wv[kb + 1];
        // (neg_a, A, neg_b, B, c_mod, C, reuse_a, reuse_b)
        cq = __builtin_amdgcn_wmma_f32_16x16x4_f32(false, a, false, bq,
                                                   (short)0, cq, false, false);
        ck = __builtin_amdgcn_wmma_f32_16x16x4_f32(false, a, false, bk,
                                                   (short)0, ck, false, false);
        cv = __builtin_amdgcn_wmma_f32_16x16x4_f32(false, a, false, bv,
                                                   (short)0, cv, false, false);
    }

    // C/D layout: lane<16 -> N=lane, M=vgpr; lane>=16 -> N=lane-16, M=vgpr+8
    // q/k/v workspace (153.6 MB total) fits MI455X's 192 MB L2: store with
    // default RT policy so the edge-phase random gathers hit L2.
    const size_t colOff = (size_t)head * HEAD_DIM + (size_t)ntile * 16 + r;
    float* qo = q + colOff;
    float* ko = k + colOff;
    float* vo = v + colOff;
#pragma unroll
    for (int m = 0; m < 8; ++m) {
        const size_t row = (size_t)(mtile * 16 + m + kh * 8) * NFEAT;
        qo[row] = cq[m];
        ko[row] = ck[m];
        vo[row] = cv[m];
    }
}

// ---------------------------------------------------------------------------
// Kernel 2: zero the output (scatter-add target). 12.8M floats = 3.2M v4f.
// ---------------------------------------------------------------------------
__global__ void __launch_bounds__(256)
zero_out(v4f* __restrict__ y)
{
    const int idx = blockIdx.x * 256 + threadIdx.x;
    y[idx] = (v4f){0.f, 0.f, 0.f, 0.f};
}

// ---------------------------------------------------------------------------
// Kernel 3: one wave32 per edge. Lane L owns features [8L, 8L+8) -> a single
// head per 8-lane subgroup. Per-head dot reduced via shfl_xor (wave32), then
// alpha * v[idx_j] scatter-added with f32 atomics (resolved in L2).
//
// Cache policy (192 MB L2 roofline reasoning):
//   - q/k/v gathers: default RT -> served from L2 (working set 153.6 MB).
//   - w_ij (819 MB), phi, idx: streamed exactly once -> non-temporal loads
//     so they don't evict the q/k/v working set from L2.
// ---------------------------------------------------------------------------
__global__ void __launch_bounds__(256)
edge_attn(const float* __restrict__ q,
          const float* __restrict__ k,
          const float* __restrict__ v,
          const float* __restrict__ w_ij,
          const int*   __restrict__ idx_i,
          const int*   __restrict__ idx_j,
          const float* __restrict__ phi,
          float* __restrict__ y)
{
    const int lane = threadIdx.x & 31;
    const int e    = blockIdx.x * 8 + (threadIdx.x >> 5);   // 0..799999

    const int i = __builtin_nontemporal_load(idx_i + e);
    const int j = __builtin_nontemporal_load(idx_j + e);

    const v4f* qi = (const v4f*)(q    + (size_t)i * NFEAT) + lane * 2;
    const v4f* kj = (const v4f*)(k    + (size_t)j * NFEAT) + lane * 2;
    const v4f* vj = (const v4f*)(v    + (size_t)j * NFEAT) + lane * 2;
    const v4f* we = (const v4f*)(w_ij + (size_t)e * NFEAT) + lane * 2;

    // single-use stream: non-temporal (global_load_b128 th:NT)
    const v4f w0 = __builtin_nontemporal_load(we + 0);
    const v4f w1 = __builtin_nontemporal_load(we + 1);
    // L2-resident working set: regular temporal
    const v4f q0 = qi[0], q1 = qi[1];
    const v4f k0 = kj[0], k1 = kj[1];
    const v4f v0 = vj[0], v1 = vj[1];

    float s = q0.x * w0.x * k0.x + q0.y * w0.y * k0.y
            + q0.z * w0.z * k0.z + q0.w * w0.w * k0.w
            + q1.x * w1.x * k1.x + q1.y * w1.y * k1.y
            + q1.z * w1.z * k1.z + q1.w * w1.w * k1.w;

    // reduce across the 8-lane subgroup that shares this head
    s += __shfl_xor(s, 1, 32);
    s += __shfl_xor(s, 2, 32);
    s += __shfl_xor(s, 4, 32);

    const float phi_e = __builtin_nontemporal_load(phi + e);
    const float alpha = s * 0.125f * phi_e;     // 1/sqrt(64) = 0.125

    float* yo = y + (size_t)i * NFEAT + lane * 8;
    atomicAdd(yo + 0, alpha * v0.x);
    atomicAdd(yo + 1, alpha * v0.y);
    atomicAdd(yo + 2, alpha * v0.z);
    atomicAdd(yo + 3, alpha * v0.w);
    atomicAdd(yo + 4, alpha * v1.x);
    atomicAdd(yo + 5, alpha * v1.y);
    atomicAdd(yo + 6, alpha * v1.z);
    atomicAdd(yo + 7, alpha * v1.w);
}

// ---------------------------------------------------------------------------
extern "C" void kernel_launch(void* const* d_in, const int* in_sizes, int n_in,
                              void* d_out, int out_size, void* d_ws, size_t ws_size,
                              hipStream_t stream)
{
    const float* x    = (const float*)d_in[0];
    const float* w_ij = (const float*)d_in[1];
    const int*   ii   = (const int*)  d_in[2];
    const int*   jj   = (const int*)  d_in[3];
    const float* phi  = (const float*)d_in[4];
    const float* Wq   = (const float*)d_in[5];
    const float* Wk   = (const float*)d_in[6];
    const float* Wv   = (const float*)d_in[7];

    float* y = (float*)d_out;

    // workspace: q | k | v, each N_NODES*NFEAT floats (51.2 MB each)
    float* q = (float*)d_ws;
    float* k = q + (size_t)N_NODES * NFEAT;
    float* v = k + (size_t)N_NODES * NFEAT;

    // 50000 waves / 8 waves-per-block
    qkv_proj_wmma<<<6250, 256, 0, stream>>>(x, Wq, Wk, Wv, q, k, v);

    // 50000*256 / (256 threads * 4 floats) = 12500 blocks
    zero_out<<<12500, 256, 0, stream>>>((v4f*)y);

    // 800000 edges, 1 wave each, 8 waves/block
    edge_attn<<<100000, 256, 0, stream>>>(q, k, v, w_ij, ii, jj, phi, y);
}